// WeightLoss_28458453303914
// MI455X (gfx1250) — compile-verified
//
#include <hip/hip_runtime.h>
#include <math.h>
#include <stdint.h>

#define NROWS 4096
#define DIM   512
#define ALPHA_C 10.0f
#define BETA_C  2.0f
#define BASE_C  0.5f
#define MARGIN_C 0.1f

typedef float v2f __attribute__((ext_vector_type(2)));
typedef float v8f __attribute__((ext_vector_type(8)));

#define BI 128            // rows per block
#define BJ 64             // cols per j-step
#define BK 64             // k tile
#define NKB (DIM / BK)    // 8 k-stages
#define LDSS (BK + 4)     // padded LDS stride (68 floats, keeps 16B row alignment)

// ---- gfx1250 async global->LDS copy (no VGPR staging, tracked by ASYNCcnt) ----
__device__ __forceinline__ void async_copy_b128(uint32_t lds_byte_addr,
                                                const float* gptr) {
  asm volatile("global_load_async_to_lds_b128 %0, %1, off"
               :
               : "v"(lds_byte_addr), "v"((uint64_t)(uintptr_t)gptr)
               : "memory");
}
__device__ __forceinline__ void wait_async_le12() {
  asm volatile("s_wait_asynccnt 0xc" ::: "memory");
}
__device__ __forceinline__ void wait_async_0() {
  asm volatile("s_wait_asynccnt 0x0" ::: "memory");
}

// Issue the 12 async b128 copies filling one (A,B) LDS stage for this thread.
__device__ __forceinline__ void fill_tiles_async(
    const float* __restrict__ X, int iblock, int jb, int kbase, int tid,
    float (*As)[LDSS], float (*Bs)[LDSS]) {
#pragma unroll
  for (int i = 0; i < 8; ++i) {          // A: 128 x 64 floats
    int e = tid + i * 256;               // 0..2047
    int row = e >> 4;                    // 0..127
    int kq = (e & 15) << 2;              // 0..60 step 4
    async_copy_b128((uint32_t)(uintptr_t)&As[row][kq],
                    &X[(size_t)(iblock + row) * DIM + kbase + kq]);
  }
#pragma unroll
  for (int i = 0; i < 4; ++i) {          // B: 64 x 64 floats
    int e = tid + i * 256;               // 0..1023
    int row = e >> 4;                    // 0..63
    int kq = (e & 15) << 2;
    async_copy_b128((uint32_t)(uintptr_t)&Bs[row][kq],
                    &X[(size_t)(jb + row) * DIM + kbase + kq]);
  }
}

// PHASE 0: per-row pos_min / neg_max.  PHASE 1: hard-mined exp sums + row losses.
template <int PHASE>
__global__ __launch_bounds__(256) void ms_sim_kernel(
    const float* __restrict__ X, const int* __restrict__ T,
    float* __restrict__ pos_min, float* __restrict__ neg_max,
    float* __restrict__ accums)
{
  __shared__ __align__(16) float As[2][BI][LDSS];   // double-buffered
  __shared__ __align__(16) float Bs[2][BJ][LDSS];

  const int tid  = threadIdx.x;
  const int wave = tid >> 5;
  const int lane = tid & 31;
  const int hl   = lane >> 4;      // half-wave select
  const int l16  = lane & 15;
  const int iblock  = blockIdx.x * BI;
  const int rowbase = iblock + wave * 16 + hl * 8;   // lane's 8 rows

  int tr[8];
#pragma unroll
  for (int r = 0; r < 8; ++r) tr[r] = T[rowbase + r];

  float pm[8], nm[8];
#pragma unroll
  for (int r = 0; r < 8; ++r) { pm[r] = 0.0f; nm[r] = 0.0f; }
  if (PHASE == 1) {
#pragma unroll
    for (int r = 0; r < 8; ++r) {
      pm[r] = pos_min[rowbase + r];
      nm[r] = neg_max[rowbase + r];
    }
  }

  float minp[8], maxn[8], psum[8], nsum[8];
#pragma unroll
  for (int r = 0; r < 8; ++r) {
    minp[r] = INFINITY; maxn[r] = -INFINITY; psum[r] = 0.0f; nsum[r] = 0.0f;
  }

  for (int jb = 0; jb < NROWS; jb += BJ) {
    v8f acc[4];
#pragma unroll
    for (int ct = 0; ct < 4; ++ct)
#pragma unroll
      for (int e = 0; e < 8; ++e) acc[ct][e] = 0.0f;

    // software pipeline: prefetch stage 0, then overlap copy(k+1) with wmma(k)
    fill_tiles_async(X, iblock, jb, 0, tid, As[0], Bs[0]);

    for (int kb = 0; kb < NKB; ++kb) {
      const int cur = kb & 1;
      if (kb + 1 < NKB) {
        // other buffer was freed by previous iteration's trailing barrier
        fill_tiles_async(X, iblock, jb, (kb + 1) * BK, tid,
                         As[1 - cur], Bs[1 - cur]);
        wait_async_le12();     // stage kb's 12 copies complete, kb+1 in flight
      } else {
        wait_async_0();
      }
      __syncthreads();         // everyone's stage-kb data resident in LDS

#pragma unroll
      for (int k0 = 0; k0 < BK; k0 += 4) {
        // A fragment 16x4 fp32: m = l16, K pair {0,1}/{2,3} by half-wave
        float2 a2 = *(const float2*)&As[cur][wave * 16 + l16][k0 + hl * 2];
        v2f a; a.x = a2.x; a.y = a2.y;
#pragma unroll
        for (int ct = 0; ct < 4; ++ct) {
          float2 b2 = *(const float2*)&Bs[cur][ct * 16 + l16][k0 + hl * 2];
          v2f b; b.x = b2.x; b.y = b2.y;
          acc[ct] = __builtin_amdgcn_wmma_f32_16x16x4_f32(
              false, a, false, b, (short)0, acc[ct], false, false);
        }
      }
      __syncthreads();         // stage-kb buffer free for reuse
    }

    // Epilogue over this wave's 16x64 sim strip.
    // C layout: VGPR r -> row (r, lanes 0-15)/(r+8, lanes 16-31), col = l16.
#pragma unroll
    for (int ct = 0; ct < 4; ++ct) {
      int col = jb + ct * 16 + l16;
      int tc  = T[col];
#pragma unroll
      for (int r = 0; r < 8; ++r) {
        float s  = acc[ct][r];
        int row  = rowbase + r;
        bool same = (tr[r] == tc);
        if (PHASE == 0) {
          if (same && row != col) minp[r] = fminf(minp[r], s);
          if (!same)              maxn[r] = fmaxf(maxn[r], s);
        } else {
          if (!same && (s + MARGIN_C > pm[r]))
            nsum[r] += __expf(ALPHA_C * (s - BASE_C));
          if (same && row != col && (s - MARGIN_C < nm[r]))
            psum[r] += __expf(-BETA_C * (s - BASE_C));
        }
      }
    }
  }

  // Half-wave (16-lane) reductions: each row's partials live in 16 lanes.
#pragma unroll
  for (int r = 0; r < 8; ++r) {
#pragma unroll
    for (int off = 8; off >= 1; off >>= 1) {
      if (PHASE == 0) {
        minp[r] = fminf(minp[r], __shfl_xor(minp[r], off, 32));
        maxn[r] = fmaxf(maxn[r], __shfl_xor(maxn[r], off, 32));
      } else {
        psum[r] += __shfl_xor(psum[r], off, 32);
        nsum[r] += __shfl_xor(nsum[r], off, 32);
      }
    }
  }

  if (l16 == 0) {  // lanes 0 and 16 own rows rowbase..rowbase+7
#pragma unroll
    for (int r = 0; r < 8; ++r) {
      int row = rowbase + r;
      if (PHASE == 0) {
        pos_min[row] = minp[r];
        neg_max[row] = maxn[r];
      } else {
        bool valid = (nm[r] + MARGIN_C > pm[r]);  // == any(hard_neg)&&any(hard_pos)
        if (valid) {
          float row_loss = (2.0f / BETA_C)  * log1pf(psum[r]) +
                           (2.0f / ALPHA_C) * log1pf(nsum[r]);
          atomicAdd(&accums[0], row_loss);
        }
      }
    }
  }
}

__global__ __launch_bounds__(256) void last_row_kernel(
    const float* __restrict__ X, const int* __restrict__ T,
    float* __restrict__ accums)
{
  int j = blockIdx.x * blockDim.x + threadIdx.x;
  const float* xl = X + (size_t)(NROWS - 1) * DIM;
  const float* xj = X + (size_t)j * DIM;
  float s = 0.0f;
  for (int k = 0; k < DIM; k += 4) {
    float4 a = *(const float4*)&xl[k];
    float4 b = *(const float4*)&xj[k];
    s += a.x * b.x + a.y * b.y + a.z * b.z + a.w * b.w;
  }
  int tlast = T[NROWS - 1];
  bool same = (T[j] == tlast);
  bool self = (j == NROWS - 1);
  float ps = (same && !self) ? s : 0.0f;
  float pc = (same && !self) ? 1.0f : 0.0f;
  float ns = (!same) ? s : 0.0f;
  float nc = (!same) ? 1.0f : 0.0f;
#pragma unroll
  for (int off = 16; off >= 1; off >>= 1) {
    ps += __shfl_xor(ps, off, 32);
    pc += __shfl_xor(pc, off, 32);
    ns += __shfl_xor(ns, off, 32);
    nc += __shfl_xor(nc, off, 32);
  }
  if ((threadIdx.x & 31) == 0) {
    atomicAdd(&accums[1], ps);
    atomicAdd(&accums[2], pc);
    atomicAdd(&accums[3], ns);
    atomicAdd(&accums[4], nc);
  }
}

__global__ void init_kernel(float* accums) {
  if (threadIdx.x < 8) accums[threadIdx.x] = 0.0f;
}

__global__ __launch_bounds__(256) void finalize_kernel(
    const float* __restrict__ pos_min, const float* __restrict__ neg_max,
    const float* __restrict__ accums, float* __restrict__ out)
{
  __shared__ int cnt_s[256];
  int t = threadIdx.x;
  int cnt = 0;
  for (int i = t; i < NROWS; i += 256) {
    bool valid = (neg_max[i] + MARGIN_C > pos_min[i]);
    cnt += valid ? 0 : 1;
  }
  cnt_s[t] = cnt;
  __syncthreads();
  for (int s = 128; s > 0; s >>= 1) {
    if (t < s) cnt_s[t] += cnt_s[t + s];
    __syncthreads();
  }
  if (t == 0) {
    out[0] = accums[1] / accums[2];                    // mean_pos_sim (last row)
    out[1] = accums[3] / accums[4];                    // mean_neg_sim (last row)
    out[2] = (float)cnt_s[0] / (float)NROWS;           // prec (fraction skipped)
    out[3] = accums[0] / (float)NROWS;                 // loss
  }
}

extern "C" void kernel_launch(void* const* d_in, const int* in_sizes, int n_in,
                              void* d_out, int out_size, void* d_ws, size_t ws_size,
                              hipStream_t stream) {
  const float* X = (const float*)d_in[0];
  const int*   T = (const int*)d_in[1];
  float* ws      = (float*)d_ws;
  float* pos_min = ws;               // 4096
  float* neg_max = ws + NROWS;       // 4096
  float* accums  = ws + 2 * NROWS;   // [0]=loss, [1..4]=last-row sums/counts
  float* out     = (float*)d_out;

  hipLaunchKernelGGL(init_kernel, dim3(1), dim3(64), 0, stream, accums);
  hipLaunchKernelGGL((ms_sim_kernel<0>), dim3(NROWS / BI), dim3(256), 0, stream,
                     X, T, pos_min, neg_max, accums);
  hipLaunchKernelGGL((ms_sim_kernel<1>), dim3(NROWS / BI), dim3(256), 0, stream,
                     X, T, pos_min, neg_max, accums);
  hipLaunchKernelGGL(last_row_kernel, dim3(NROWS / 256), dim3(256), 0, stream,
                     X, T, accums);
  hipLaunchKernelGGL(finalize_kernel, dim3(1), dim3(256), 0, stream,
                     pos_min, neg_max, accums, out);
}